// CliffordSpectralConv3d_73005854098018
// MI455X (gfx1250) — compile-verified
//
#include <hip/hip_runtime.h>
#include <hip/hip_bf16.h>

typedef float v2f __attribute__((ext_vector_type(2)));
typedef float v8f __attribute__((ext_vector_type(8)));

#define PI_F 3.14159265358979323846f

// D(16x16,f32) = A(16x4,f32) * B(4x16,f32) + C  -- CDNA5 V_WMMA_F32_16X16X4_F32
static __device__ inline v8f wmma4(v2f a, v2f b, v8f c) {
  return __builtin_amdgcn_wmma_f32_16x16x4_f32(false, a, false, b, (short)0, c,
                                               false, false);
}

// ---- CDNA5 async global->LDS copies (ASYNCcnt path), guarded ----
#if defined(__has_builtin)
#if __has_builtin(__builtin_amdgcn_global_load_async_to_lds_b32) && \
    __has_builtin(__builtin_amdgcn_global_load_async_to_lds_b128)
#define HAVE_ASYNC_LDS 1
#endif
#endif
#ifndef HAVE_ASYNC_LDS
#define HAVE_ASYNC_LDS 0
#endif

#if HAVE_ASYNC_LDS
typedef int vi4 __attribute__((vector_size(16)));
static __device__ inline void async_b32(const void* g, void* l) {
  __builtin_amdgcn_global_load_async_to_lds_b32(
      (__attribute__((address_space(1))) int*)(void*)g,
      (__attribute__((address_space(3))) int*)l, 0, 0);
}
static __device__ inline void async_b128(const void* g, void* l) {
  __builtin_amdgcn_global_load_async_to_lds_b128(
      (__attribute__((address_space(1))) vi4*)(void*)g,
      (__attribute__((address_space(3))) vi4*)l, 0, 0);
}
#if __has_builtin(__builtin_amdgcn_s_wait_asynccnt)
static __device__ inline void wait_async0() {
  __builtin_amdgcn_s_wait_asynccnt(0);
}
#else
static __device__ inline void wait_async0() {
  asm volatile("s_wait_asynccnt 0x0" ::: "memory");
}
#endif
#endif

// Clifford geometric-product table for signature (1,1,1), blade order
// [1,e1,e2,e3,e12,e13,e23,e123]:  e_s * e_t = GP_S[s][t] * e_{GP_R[s][t]}
static constexpr int GP_R[8][8] = {
  {0,1,2,3,4,5,6,7},
  {1,0,4,5,2,3,7,6},
  {2,4,0,6,1,7,3,5},
  {3,5,6,0,7,1,2,4},
  {4,2,1,7,0,6,5,3},
  {5,3,7,1,6,0,4,2},
  {6,7,3,2,5,4,0,1},
  {7,6,5,4,3,2,1,0}};
static constexpr float GP_S[8][8] = {
  { 1, 1, 1, 1, 1, 1, 1, 1},
  { 1, 1, 1, 1, 1, 1, 1, 1},
  { 1,-1, 1, 1,-1,-1, 1,-1},
  { 1,-1,-1, 1, 1,-1,-1, 1},
  { 1,-1, 1, 1,-1,-1, 1,-1},
  { 1,-1,-1, 1, 1,-1,-1, 1},
  { 1, 1,-1, 1,-1, 1,-1,-1},
  { 1, 1,-1, 1,-1, 1,-1,-1}};

// ---------------- sizes ----------------
// x:(2,16,48,48,48,8)  w:(8,16,16,24,24,24)  out:(2,16,48,48,48,8)
#define NB   2
#define NC   16
#define ND   48
#define NM   24          // kept modes per axis (2*12)
#define NMODE (24*24*24) // 13824

// workspace layout (floats)
#define OFF_WR   0        // 32x48 (rows 24..31 zero)
#define OFF_WI   1536
#define OFF_WIN  3072
#define OFF_VR   4608     // 48x24, scaled by 1/48
#define OFF_VI   5760
#define OFF_VIN  6912
#define OFF_TABEND 8192
#define SZ_A (2*4*16*48*48*24*2)   // 14,155,776  (F1 / I2)
#define SZ_B (2*4*16*48*24*24*2)   //  7,077,888  (F2 / I1)
#define SZ_C (2*4*16*NMODE*2)      //  3,538,944  (F3)
#define SZ_D (2*4*16*NMODE*2)      //  3,538,944  (Om)

__global__ void k_init_tables(float* tab) {
  for (int i = threadIdx.x; i < 32 * 48; i += blockDim.x) {
    int k = i / 48, n = i % 48;
    float wr = 0.f, wi = 0.f;
    if (k < 24) {
      int f = (k < 12) ? k : k + 24;
      float ang = -2.f * PI_F * (float)(f * n) / 48.f;
      wr = __cosf(ang); wi = __sinf(ang);
    }
    tab[OFF_WR + i] = wr;
    tab[OFF_WI + i] = wi;
    tab[OFF_WIN + i] = -wi;
  }
  for (int i = threadIdx.x; i < 48 * 24; i += blockDim.x) {
    int xx = i / 24, k = i % 24;
    int f = (k < 12) ? k : k + 24;
    float ang = 2.f * PI_F * (float)(f * xx) / 48.f;
    float vr = __cosf(ang) * (1.f / 48.f);
    float vi = __sinf(ang) * (1.f / 48.f);
    tab[OFF_VR + i] = vr;
    tab[OFF_VI + i] = vi;
    tab[OFF_VIN + i] = -vi;
  }
}

// ---------------- stage 1: forward DFT along z ----------------
// grid = B*C*48*3, block = 256 (8 waves: dual d 0..3  x  M-tile 0..1)
__global__ __launch_bounds__(256) void k_fwd_z(
    const float* __restrict__ x, const float* __restrict__ tab,
    float* __restrict__ f1r, float* __restrict__ f1i) {
  __shared__ __align__(16) float lds[16 * 388];  // [line(yy)][z*8+blade], padded
  int blk = blockIdx.x;
  int yt = blk % 3;
  int xx = (blk / 3) % 48;
  int bc = blk / (3 * 48);               // b*16 + c
  const float* src = x + ((size_t)(bc * 48 + xx) * 48 + yt * 16) * 384;
#if HAVE_ASYNC_LDS
  for (int i = threadIdx.x; i < 1536; i += 256) {
    int line = i / 96, w = i % 96;
    async_b128(&src[i * 4], &lds[line * 388 + w * 4]);
  }
  wait_async0();
#else
  const float4* s4 = (const float4*)src;
  for (int i = threadIdx.x; i < 1536; i += 256) {
    float4 v = s4[i];
    int line = i / 96, w = i % 96;
    *(float4*)&lds[line * 388 + w * 4] = v;
  }
#endif
  __syncthreads();
  int wave = threadIdx.x >> 5, lane = threadIdx.x & 31;
  int d = wave & 3, mt = wave >> 2;
  const float* Wr = tab + OFF_WR;
  const float* Wi = tab + OFF_WI;
  const float* Wn = tab + OFF_WIN;
  int mrow = mt * 16 + (lane & 15);
  int khalf = (lane >> 4) << 1;
  v8f accR = {0,0,0,0,0,0,0,0}, accI = {0,0,0,0,0,0,0,0};
#pragma unroll
  for (int ks = 0; ks < 12; ++ks) {
    int k0 = ks * 4 + khalf;
    v2f ar = {Wr[mrow * 48 + k0], Wr[mrow * 48 + k0 + 1]};
    v2f ai = {Wi[mrow * 48 + k0], Wi[mrow * 48 + k0 + 1]};
    v2f an = {Wn[mrow * 48 + k0], Wn[mrow * 48 + k0 + 1]};
    const float* b0 = &lds[(lane & 15) * 388 + k0 * 8];
    v2f br = {b0[d], b0[8 + d]};
    v2f bi = {b0[7 - d], b0[15 - d]};
    accR = wmma4(ar, br, accR); accR = wmma4(an, bi, accR);
    accI = wmma4(ai, br, accI); accI = wmma4(ar, bi, accI);
  }
  int b = bc / 16, c = bc % 16;
  int yy = yt * 16 + (lane & 15);
  size_t ob = (((size_t)((b * 4 + d) * 16 + c) * 48 + xx) * 48 + yy) * 24;
#pragma unroll
  for (int v = 0; v < 8; ++v) {
    int kz = mt * 16 + v + ((lane >> 4) << 3);
    if (kz < 24) { f1r[ob + kz] = accR[v]; f1i[ob + kz] = accI[v]; }
  }
}

// ---------------- stage 2: forward DFT along y ----------------
// grid = 128*48, block = 128 (4 waves: M-tile(ky) x N-tile(kz))
__global__ __launch_bounds__(128) void k_fwd_y(
    const float* __restrict__ f1r, const float* __restrict__ f1i,
    const float* __restrict__ tab, float* __restrict__ f2r,
    float* __restrict__ f2i) {
  __shared__ __align__(16) float lr[48 * 24], li[48 * 24];  // [yy][kz]
  int blk = blockIdx.x;
  int xx = blk % 48;
  int bdc = blk / 48;
  size_t off = ((size_t)bdc * 48 + xx) * 1152;
#if HAVE_ASYNC_LDS
  for (int i = threadIdx.x * 4; i < 1152; i += 128 * 4) {
    async_b128(&f1r[off + i], &lr[i]);
    async_b128(&f1i[off + i], &li[i]);
  }
  wait_async0();
#else
  for (int i = threadIdx.x; i < 1152; i += 128) {
    lr[i] = f1r[off + i]; li[i] = f1i[off + i];
  }
#endif
  __syncthreads();
  int wave = threadIdx.x >> 5, lane = threadIdx.x & 31;
  int mt = wave >> 1, nt = wave & 1;
  int kz = nt * 16 + (lane & 15);
  int kzc = (kz < 24) ? kz : 23;
  const float* Wr = tab + OFF_WR;
  const float* Wi = tab + OFF_WI;
  const float* Wn = tab + OFF_WIN;
  int mrow = mt * 16 + (lane & 15);
  int khalf = (lane >> 4) << 1;
  v8f accR = {0,0,0,0,0,0,0,0}, accI = {0,0,0,0,0,0,0,0};
#pragma unroll
  for (int ks = 0; ks < 12; ++ks) {
    int k0 = ks * 4 + khalf;
    v2f ar = {Wr[mrow * 48 + k0], Wr[mrow * 48 + k0 + 1]};
    v2f ai = {Wi[mrow * 48 + k0], Wi[mrow * 48 + k0 + 1]};
    v2f an = {Wn[mrow * 48 + k0], Wn[mrow * 48 + k0 + 1]};
    v2f br = {lr[k0 * 24 + kzc], lr[(k0 + 1) * 24 + kzc]};
    v2f bi = {li[k0 * 24 + kzc], li[(k0 + 1) * 24 + kzc]};
    accR = wmma4(ar, br, accR); accR = wmma4(an, bi, accR);
    accI = wmma4(ai, br, accI); accI = wmma4(ar, bi, accI);
  }
  size_t ob = ((size_t)bdc * 48 + xx) * 576;
#pragma unroll
  for (int v = 0; v < 8; ++v) {
    int ky = mt * 16 + v + ((lane >> 4) << 3);
    if (ky < 24 && kz < 24) {
      f2r[ob + ky * 24 + kz] = accR[v];
      f2i[ob + ky * 24 + kz] = accI[v];
    }
  }
}

// ---------------- stage 3: forward DFT along x ----------------
// grid = 128*24, block = 128
__global__ __launch_bounds__(128) void k_fwd_x(
    const float* __restrict__ f2r, const float* __restrict__ f2i,
    const float* __restrict__ tab, float* __restrict__ f3r,
    float* __restrict__ f3i) {
  __shared__ __align__(16) float lr[48 * 24], li[48 * 24];  // [xx][kz]
  int blk = blockIdx.x;
  int ky = blk % 24;
  int bdc = blk / 24;
#if HAVE_ASYNC_LDS
  for (int i = threadIdx.x; i < 1152; i += 128) {
    int n = i / 24, kz = i % 24;
    size_t off = ((size_t)bdc * 48 + n) * 576 + (size_t)ky * 24 + kz;
    async_b32(&f2r[off], &lr[i]);
    async_b32(&f2i[off], &li[i]);
  }
  wait_async0();
#else
  for (int i = threadIdx.x; i < 1152; i += 128) {
    int n = i / 24, kz = i % 24;
    size_t off = ((size_t)bdc * 48 + n) * 576 + (size_t)ky * 24 + kz;
    lr[i] = f2r[off]; li[i] = f2i[off];
  }
#endif
  __syncthreads();
  int wave = threadIdx.x >> 5, lane = threadIdx.x & 31;
  int mt = wave >> 1, nt = wave & 1;
  int kz = nt * 16 + (lane & 15);
  int kzc = (kz < 24) ? kz : 23;
  const float* Wr = tab + OFF_WR;
  const float* Wi = tab + OFF_WI;
  const float* Wn = tab + OFF_WIN;
  int mrow = mt * 16 + (lane & 15);
  int khalf = (lane >> 4) << 1;
  v8f accR = {0,0,0,0,0,0,0,0}, accI = {0,0,0,0,0,0,0,0};
#pragma unroll
  for (int ks = 0; ks < 12; ++ks) {
    int k0 = ks * 4 + khalf;
    v2f ar = {Wr[mrow * 48 + k0], Wr[mrow * 48 + k0 + 1]};
    v2f ai = {Wi[mrow * 48 + k0], Wi[mrow * 48 + k0 + 1]};
    v2f an = {Wn[mrow * 48 + k0], Wn[mrow * 48 + k0 + 1]};
    v2f br = {lr[k0 * 24 + kzc], lr[(k0 + 1) * 24 + kzc]};
    v2f bi = {li[k0 * 24 + kzc], li[(k0 + 1) * 24 + kzc]};
    accR = wmma4(ar, br, accR); accR = wmma4(an, bi, accR);
    accI = wmma4(ai, br, accI); accI = wmma4(ar, bi, accI);
  }
#pragma unroll
  for (int v = 0; v < 8; ++v) {
    int kx = mt * 16 + v + ((lane >> 4) << 3);
    if (kx < 24 && kz < 24) {
      size_t o = (size_t)bdc * NMODE + kx * 576 + ky * 24 + kz;
      f3r[o] = accR[v]; f3i[o] = accI[v];
    }
  }
}

// ---------------- stage 4: Clifford spectral mixing ----------------
// grid = 864 (16 modes per block), block = 512 (16 waves, 1 mode each)
__global__ __launch_bounds__(512) void k_mix(
    const float* __restrict__ f3r, const float* __restrict__ f3i,
    const float* __restrict__ wgt, float* __restrict__ omr,
    float* __restrict__ omi) {
  extern __shared__ float sm[];
  float* xl = sm;          // [i][m=(b*8+s)][ml pad 17] : 16*16*17 = 4352
  float* wl = sm + 4352;   // [t][i][o][ml pad 17]      : 8*16*16*17 = 34816
  int mode0 = blockIdx.x * 16;
#if HAVE_ASYNC_LDS
  for (int j = threadIdx.x; j < 4096; j += 512) {
    int ml = j & 15, i = (j >> 4) & 15, s = (j >> 8) & 7, b = j >> 11;
    const float* src =
        (s < 4) ? &f3r[((size_t)(b * 4 + s) * 16 + i) * NMODE + mode0 + ml]
                : &f3i[((size_t)(b * 4 + (7 - s)) * 16 + i) * NMODE + mode0 + ml];
    async_b32(src, &xl[(i * 16 + (b * 8 + s)) * 17 + ml]);
  }
  for (int j = threadIdx.x; j < 32768; j += 512) {
    int ml = j & 15, i = (j >> 4) & 15, o = (j >> 8) & 15, t = j >> 12;
    async_b32(&wgt[(((size_t)t * 16 + o) * 16 + i) * NMODE + mode0 + ml],
              &wl[((t * 16 + i) * 16 + o) * 17 + ml]);
  }
  wait_async0();
#else
  for (int j = threadIdx.x; j < 4096; j += 512) {
    int ml = j & 15, i = (j >> 4) & 15, s = (j >> 8) & 7, b = j >> 11;
    float v;
    if (s < 4) v = f3r[((size_t)(b * 4 + s) * 16 + i) * NMODE + mode0 + ml];
    else       v = f3i[((size_t)(b * 4 + (7 - s)) * 16 + i) * NMODE + mode0 + ml];
    xl[(i * 16 + (b * 8 + s)) * 17 + ml] = v;
  }
  for (int j = threadIdx.x; j < 32768; j += 512) {
    int ml = j & 15, i = (j >> 4) & 15, o = (j >> 8) & 15, t = j >> 12;
    wl[((t * 16 + i) * 16 + o) * 17 + ml] =
        wgt[(((size_t)t * 16 + o) * 16 + i) * NMODE + mode0 + ml];
  }
#endif
  __syncthreads();
  int wave = threadIdx.x >> 5, lane = threadIdx.x & 31;
  int ml = wave;
  int mrow = lane & 15;            // A row = (b*8+s)
  int o = lane & 15;               // B col = out channel
  int khalf = (lane >> 4) << 1;
  v2f a[4];
#pragma unroll
  for (int kk = 0; kk < 4; ++kk) {
    int i0 = kk * 4 + khalf;
    a[kk].x = xl[(i0 * 16 + mrow) * 17 + ml];
    a[kk].y = xl[((i0 + 1) * 16 + mrow) * 17 + ml];
  }
  float acc[8] = {0, 0, 0, 0, 0, 0, 0, 0};
#pragma unroll
  for (int t = 0; t < 8; ++t) {
    v8f y = {0,0,0,0,0,0,0,0};
#pragma unroll
    for (int kk = 0; kk < 4; ++kk) {
      int i0 = kk * 4 + khalf;
      v2f bf;
      bf.x = wl[((t * 16 + i0) * 16 + o) * 17 + ml];
      bf.y = wl[((t * 16 + i0 + 1) * 16 + o) * 17 + ml];
      y = wmma4(a[kk], bf, y);
    }
#pragma unroll
    for (int s = 0; s < 8; ++s) acc[GP_R[s][t]] += GP_S[s][t] * y[s];
  }
  // transpose through LDS for coalesced output stores
  __syncthreads();
  float* tr = sm;  // 4096 floats: [b][r][o][ml]
  int b = lane >> 4;
#pragma unroll
  for (int r = 0; r < 8; ++r) tr[((b * 8 + r) * 16 + o) * 16 + ml] = acc[r];
  __syncthreads();
  for (int j = threadIdx.x; j < 4096; j += 512) {
    int ml2 = j & 15, o2 = (j >> 4) & 15, r = (j >> 8) & 7, b2 = j >> 11;
    float v = tr[j];
    size_t mode = mode0 + ml2;
    if (r < 4) omr[((size_t)(b2 * 4 + r) * 16 + o2) * NMODE + mode] = v;
    else       omi[((size_t)(b2 * 4 + (7 - r)) * 16 + o2) * NMODE + mode] = v;
  }
}

// ---------------- stage 5: inverse DFT along x (with scatter quirk) ----
// grid = 128*24, block = 192 (6 waves: 3 M-tiles(xx) x 2 N-tiles(kz))
__global__ __launch_bounds__(192) void k_inv_x(
    const float* __restrict__ omr, const float* __restrict__ omi,
    const float* __restrict__ tab, float* __restrict__ i1r,
    float* __restrict__ i1i) {
  __shared__ __align__(16) float lr[576], li[576];  // [kx][kz]
  int blk = blockIdx.x;
  int ky = blk % 24;
  int bdo = blk / 24;
#if HAVE_ASYNC_LDS
  for (int i = threadIdx.x; i < 576; i += 192) {
    int kx = i / 24, kz = i % 24;
    size_t off = (size_t)bdo * NMODE + kx * 576 + ky * 24 + kz;
    async_b32(&omr[off], &lr[i]);
    async_b32(&omi[off], &li[i]);
  }
  wait_async0();
#else
  for (int i = threadIdx.x; i < 576; i += 192) {
    int kx = i / 24, kz = i % 24;
    size_t off = (size_t)bdo * NMODE + kx * 576 + ky * 24 + kz;
    lr[i] = omr[off]; li[i] = omi[off];
  }
#endif
  __syncthreads();
  int wave = threadIdx.x >> 5, lane = threadIdx.x & 31;
  int mt = wave >> 1, nt = wave & 1;
  int kz = nt * 16 + (lane & 15);
  int kzc = (kz < 24) ? kz : 23;
  // upstream quirk: out_ft[-m1:, :m2, -m3:] takes om[:m1] rows
  bool region = (ky < 12) && (kzc >= 12);
  const float* Vr = tab + OFF_VR;
  const float* Vi = tab + OFF_VI;
  const float* Vn = tab + OFF_VIN;
  int mrow = mt * 16 + (lane & 15);
  int khalf = (lane >> 4) << 1;
  v8f accR = {0,0,0,0,0,0,0,0}, accI = {0,0,0,0,0,0,0,0};
#pragma unroll
  for (int ks = 0; ks < 6; ++ks) {
    int k0 = ks * 4 + khalf;
    v2f ar = {Vr[mrow * 24 + k0], Vr[mrow * 24 + k0 + 1]};
    v2f ai = {Vi[mrow * 24 + k0], Vi[mrow * 24 + k0 + 1]};
    v2f an = {Vn[mrow * 24 + k0], Vn[mrow * 24 + k0 + 1]};
    int n0 = k0, n1 = k0 + 1;
    if (region) { if (n0 >= 12) n0 -= 12; if (n1 >= 12) n1 -= 12; }
    v2f br = {lr[n0 * 24 + kzc], lr[n1 * 24 + kzc]};
    v2f bi = {li[n0 * 24 + kzc], li[n1 * 24 + kzc]};
    accR = wmma4(ar, br, accR); accR = wmma4(an, bi, accR);
    accI = wmma4(ai, br, accI); accI = wmma4(ar, bi, accI);
  }
#pragma unroll
  for (int v = 0; v < 8; ++v) {
    int xx = mt * 16 + v + ((lane >> 4) << 3);
    if (kz < 24) {
      size_t o = ((size_t)bdo * 48 + xx) * 576 + ky * 24 + kz;
      i1r[o] = accR[v]; i1i[o] = accI[v];
    }
  }
}

// ---------------- stage 6: inverse DFT along y ----------------
// grid = 128*48, block = 192
__global__ __launch_bounds__(192) void k_inv_y(
    const float* __restrict__ i1r, const float* __restrict__ i1i,
    const float* __restrict__ tab, float* __restrict__ i2r,
    float* __restrict__ i2i) {
  __shared__ __align__(16) float lr[576], li[576];  // [ky][kz]
  int blk = blockIdx.x;
  int xx = blk % 48;
  int bdo = blk / 48;
  size_t off = ((size_t)bdo * 48 + xx) * 576;
#if HAVE_ASYNC_LDS
  for (int i = threadIdx.x * 4; i < 576; i += 192 * 4) {
    async_b128(&i1r[off + i], &lr[i]);
    async_b128(&i1i[off + i], &li[i]);
  }
  wait_async0();
#else
  for (int i = threadIdx.x; i < 576; i += 192) {
    lr[i] = i1r[off + i]; li[i] = i1i[off + i];
  }
#endif
  __syncthreads();
  int wave = threadIdx.x >> 5, lane = threadIdx.x & 31;
  int mt = wave >> 1, nt = wave & 1;
  int kz = nt * 16 + (lane & 15);
  int kzc = (kz < 24) ? kz : 23;
  const float* Vr = tab + OFF_VR;
  const float* Vi = tab + OFF_VI;
  const float* Vn = tab + OFF_VIN;
  int mrow = mt * 16 + (lane & 15);
  int khalf = (lane >> 4) << 1;
  v8f accR = {0,0,0,0,0,0,0,0}, accI = {0,0,0,0,0,0,0,0};
#pragma unroll
  for (int ks = 0; ks < 6; ++ks) {
    int k0 = ks * 4 + khalf;
    v2f ar = {Vr[mrow * 24 + k0], Vr[mrow * 24 + k0 + 1]};
    v2f ai = {Vi[mrow * 24 + k0], Vi[mrow * 24 + k0 + 1]};
    v2f an = {Vn[mrow * 24 + k0], Vn[mrow * 24 + k0 + 1]};
    v2f br = {lr[k0 * 24 + kzc], lr[(k0 + 1) * 24 + kzc]};
    v2f bi = {li[k0 * 24 + kzc], li[(k0 + 1) * 24 + kzc]};
    accR = wmma4(ar, br, accR); accR = wmma4(an, bi, accR);
    accI = wmma4(ai, br, accI); accI = wmma4(ar, bi, accI);
  }
#pragma unroll
  for (int v = 0; v < 8; ++v) {
    int yy = mt * 16 + v + ((lane >> 4) << 3);
    if (kz < 24) {
      size_t o = (((size_t)bdo * 48 + xx) * 48 + yy) * 24 + kz;
      i2r[o] = accR[v]; i2i[o] = accI[v];
    }
  }
}

// ---------------- stage 7: inverse DFT along z + blade interleave ------
// grid = B*Co*48*3, block = 384 (12 waves: dual d 0..3 x M-tile 0..2)
__global__ __launch_bounds__(384) void k_inv_z(
    const float* __restrict__ i2r, const float* __restrict__ i2i,
    const float* __restrict__ tab, float* __restrict__ out) {
  __shared__ __align__(16) float lds[8 * 400];  // [(d*2+comp)][yy][kz pad 25]
  int blk = blockIdx.x;
  int yt = blk % 3;
  int t1 = blk / 3;
  int xx = t1 % 48;
  int bo = t1 / 48;               // b*16 + o
  int b = bo / 16, o = bo % 16;
#if HAVE_ASYNC_LDS
  for (int i = threadIdx.x; i < 3072; i += 384) {
    int kz = i % 24;
    int yy = (i / 24) % 16;
    int comp = (i / 384) & 1;
    int d = i / 768;
    size_t off = (((size_t)((b * 4 + d) * 16 + o) * 48 + xx) * 48 + (yt * 16 + yy)) * 24 + kz;
    const float* src = comp ? &i2i[off] : &i2r[off];
    async_b32(src, &lds[(d * 2 + comp) * 400 + yy * 25 + kz]);
  }
  wait_async0();
#else
  for (int i = threadIdx.x; i < 3072; i += 384) {
    int kz = i % 24;
    int yy = (i / 24) % 16;
    int comp = (i / 384) & 1;
    int d = i / 768;
    size_t off = (((size_t)((b * 4 + d) * 16 + o) * 48 + xx) * 48 + (yt * 16 + yy)) * 24 + kz;
    lds[(d * 2 + comp) * 400 + yy * 25 + kz] = comp ? i2i[off] : i2r[off];
  }
#endif
  __syncthreads();
  int wave = threadIdx.x >> 5, lane = threadIdx.x & 31;
  int d = wave & 3, mt = wave >> 2;
  const float* Vr = tab + OFF_VR;
  const float* Vi = tab + OFF_VI;
  const float* Vn = tab + OFF_VIN;
  const float* Lr = &lds[(d * 2 + 0) * 400];
  const float* Li = &lds[(d * 2 + 1) * 400];
  int mrow = mt * 16 + (lane & 15);
  int khalf = (lane >> 4) << 1;
  int line = lane & 15;
  v8f accR = {0,0,0,0,0,0,0,0}, accI = {0,0,0,0,0,0,0,0};
#pragma unroll
  for (int ks = 0; ks < 6; ++ks) {
    int k0 = ks * 4 + khalf;
    v2f ar = {Vr[mrow * 24 + k0], Vr[mrow * 24 + k0 + 1]};
    v2f ai = {Vi[mrow * 24 + k0], Vi[mrow * 24 + k0 + 1]};
    v2f an = {Vn[mrow * 24 + k0], Vn[mrow * 24 + k0 + 1]};
    v2f br = {Lr[line * 25 + k0], Lr[line * 25 + k0 + 1]};
    v2f bi = {Li[line * 25 + k0], Li[line * 25 + k0 + 1]};
    accR = wmma4(ar, br, accR); accR = wmma4(an, bi, accR);
    accI = wmma4(ai, br, accI); accI = wmma4(ar, bi, accI);
  }
  int yy = yt * 16 + (lane & 15);
  size_t ob = (((size_t)(b * 16 + o) * 48 + xx) * 48 + yy) * 48;
#pragma unroll
  for (int v = 0; v < 8; ++v) {
    int zz = mt * 16 + v + ((lane >> 4) << 3);
    out[(ob + zz) * 8 + d] = accR[v];       // blade d     = Re(ifft dual d)
    out[(ob + zz) * 8 + 7 - d] = accI[v];   // blade 7-d   = Im(ifft dual d)
  }
}

extern "C" void kernel_launch(void* const* d_in, const int* in_sizes, int n_in,
                              void* d_out, int out_size, void* d_ws,
                              size_t ws_size, hipStream_t stream) {
  const float* x = (const float*)d_in[0];
  const float* w = (const float*)d_in[1];
  float* out = (float*)d_out;
  float* ws = (float*)d_ws;

  float* tab = ws;
  float* bufA = ws + OFF_TABEND;             // F1 / I2
  float* bufB = bufA + SZ_A;                 // F2 / I1
  float* bufC = bufB + SZ_B;                 // F3
  float* bufD = bufC + SZ_C;                 // Om

  float* f1r = bufA;           float* f1i = bufA + SZ_A / 2;
  float* f2r = bufB;           float* f2i = bufB + SZ_B / 2;
  float* f3r = bufC;           float* f3i = bufC + SZ_C / 2;
  float* omr = bufD;           float* omi = bufD + SZ_D / 2;
  float* i1r = f2r;            float* i1i = f2i;   // alias (F2 dead)
  float* i2r = f1r;            float* i2i = f1i;   // alias (F1 dead)

  k_init_tables<<<1, 256, 0, stream>>>(tab);
  k_fwd_z<<<NB * NC * 48 * 3, 256, 0, stream>>>(x, tab, f1r, f1i);
  k_fwd_y<<<128 * 48, 128, 0, stream>>>(f1r, f1i, tab, f2r, f2i);
  k_fwd_x<<<128 * 24, 128, 0, stream>>>(f2r, f2i, tab, f3r, f3i);
  k_mix<<<NMODE / 16, 512, (4352 + 34816) * sizeof(float), stream>>>(
      f3r, f3i, w, omr, omi);
  k_inv_x<<<128 * 24, 192, 0, stream>>>(omr, omi, tab, i1r, i1i);
  k_inv_y<<<128 * 48, 192, 0, stream>>>(i1r, i1i, tab, i2r, i2i);
  k_inv_z<<<NB * NC * 48 * 3, 384, 0, stream>>>(i2r, i2i, tab, out);
}